// QLSTM_65481071406401
// MI455X (gfx1250) — compile-verified
//
#include <hip/hip_runtime.h>

typedef __attribute__((ext_vector_type(2))) float v2f;
typedef __attribute__((ext_vector_type(4))) float v4f;
typedef __attribute__((ext_vector_type(8))) float v8f;

#define SEQ 512
#define BATCH 256
#define IN_DIM 128
#define HID 256
#define KTOT 384  // IN_DIM + HID

// ---- LDS layout (floats) ----
#define H_STRIDE 258                       // pad: bank-conflict-free, 8B aligned float2
#define H_OFF 0
#define H_FLOATS (BATCH * H_STRIDE)        // 66048
#define WT_STRIDE 25                       // cols 0..23 real, col 24 = zero pad
#define WT_OFF H_FLOATS
#define WT_FLOATS (KTOT * WT_STRIDE)       // 9600
#define Z_STRIDE 24
#define Z_OFF (WT_OFF + WT_FLOATS)         // 75648 (16B aligned in bytes)
#define Z_FLOATS (BATCH * Z_STRIDE)        // 6144
#define LDS_BYTES ((Z_OFF + Z_FLOATS) * 4) // 327168 <= 327680

__device__ __forceinline__ float sigmoidf_(float x) {
  return 1.0f / (1.0f + __expf(-x));
}

__global__ __launch_bounds__(1024, 1)
void qlstm_persistent(const float* __restrict__ inp,
                      const float* __restrict__ Wf, const float* __restrict__ bf,
                      const float* __restrict__ Wi, const float* __restrict__ bi,
                      const float* __restrict__ Wu, const float* __restrict__ bu,
                      const float* __restrict__ Wo, const float* __restrict__ bo,
                      const float* __restrict__ Wh, const float* __restrict__ bh,
                      float* __restrict__ out, float* __restrict__ cstate) {
  extern __shared__ float lds[];
  const int T = threadIdx.x;
  const int lane = T & 31;
  const int wave = T >> 5;
  const int lhalf = lane >> 4;   // 0/1
  const int lmod = lane & 15;

  // ---- GEMM1 (WMMA f32 16x16x4) per-wave constants: tile = (mtile, ntile) ----
  const int mtile = wave >> 1;               // 0..15  (16 batch rows each)
  const int ntile = wave & 1;                // 0..1   (16 cols each; valid cols < 24)
  const int b0 = mtile * 16;
  const int nCol = ntile * 16 + lmod;        // 0..31
  const int wtCol = (nCol < 24) ? nCol : 24; // col 24 holds zeros
  float biasv = 0.f;
  if (nCol < 24) {
    const int g = nCol / 6, w = nCol % 6;
    const float* bg = (g == 0) ? bf : (g == 1) ? bi : (g == 2) ? bu : bo;
    biasv = bg[w];
  }

  // ---- GEMM2 + LSTM elementwise constants: thread owns 16 b-rows x 4 n-cols ----
  const int bblk = T >> 6;        // 0..15
  const int n0 = (T & 63) * 4;    // 0..252
  float wh[4][6];
  float bhv[4];
#pragma unroll
  for (int j = 0; j < 4; ++j) {
    bhv[j] = bh[n0 + j];
#pragma unroll
    for (int w = 0; w < 6; ++w) wh[j][w] = Wh[(n0 + j) * 6 + w];
  }

  // ---- one-time LDS init: zero h0, stage W_all^T (K x 24, zero-padded col) ----
  for (int i = T; i < H_FLOATS; i += 1024) lds[H_OFF + i] = 0.f;
  for (int i = T; i < WT_FLOATS; i += 1024) {
    const int k = i / WT_STRIDE, n = i % WT_STRIDE;
    float v = 0.f;
    if (n < 24) {
      const int g = n / 6, w = n % 6;
      const float* Wg = (g == 0) ? Wf : (g == 1) ? Wi : (g == 2) ? Wu : Wo;
      v = Wg[w * KTOT + k];  // Wt[k][n] = W_g[w][k]
    }
    lds[WT_OFF + i] = v;
  }
  __syncthreads();

  for (int t = 0; t < SEQ; ++t) {
    // ============ GEMM1: Z = [x_t, h] @ W_all^T + bias  (f32 WMMA) ============
    v8f acc;
#pragma unroll
    for (int r = 0; r < 8; ++r) acc[r] = biasv;

    // x part of K (from global), A-frag: lane -> (M=lmod, K=k0+2*lhalf..+1)
    const float* xrow =
        inp + ((size_t)t * BATCH + b0 + lmod) * IN_DIM + 2 * lhalf;
#pragma unroll 8
    for (int k0 = 0; k0 < IN_DIM; k0 += 4) {
      v2f a2 = *(const v2f*)(xrow + k0);
      const int kb = k0 + 2 * lhalf;
      v2f b2;
      b2[0] = lds[WT_OFF + kb * WT_STRIDE + wtCol];
      b2[1] = lds[WT_OFF + (kb + 1) * WT_STRIDE + wtCol];
      acc = __builtin_amdgcn_wmma_f32_16x16x4_f32(false, a2, false, b2,
                                                  (short)0, acc, false, false);
    }
    // h part of K (from LDS)
    const int hrow = H_OFF + (b0 + lmod) * H_STRIDE + 2 * lhalf;
#pragma unroll 8
    for (int k0 = 0; k0 < HID; k0 += 4) {
      v2f a2 = *(const v2f*)&lds[hrow + k0];
      const int kb = IN_DIM + k0 + 2 * lhalf;
      v2f b2;
      b2[0] = lds[WT_OFF + kb * WT_STRIDE + wtCol];
      b2[1] = lds[WT_OFF + (kb + 1) * WT_STRIDE + wtCol];
      acc = __builtin_amdgcn_wmma_f32_16x16x4_f32(false, a2, false, b2,
                                                  (short)0, acc, false, false);
    }
    // scatter D tile to Z: lane holds (M = r + 8*lhalf, N = lmod)
    if (nCol < 24) {
#pragma unroll
      for (int r = 0; r < 8; ++r)
        lds[Z_OFF + (b0 + r + 8 * lhalf) * Z_STRIDE + nCol] = acc[r];
    }
    __syncthreads();

    // ============ qgate, closed form: <Z_w> = prod of cosines ============
    {
      const int b = T & 255;
      const int g = T >> 8;                      // 0..3 (gate)
      const int zr = Z_OFF + b * Z_STRIDE + g * 6;
      const float c0 = __cosf(lds[zr + 0]);
      const float c1 = __cosf(lds[zr + 1]);
      const float c2 = __cosf(lds[zr + 2]);
      const float c3 = __cosf(lds[zr + 3]);
      const float c4 = __cosf(lds[zr + 4]);
      const float c5 = __cosf(lds[zr + 5]);
      const float t12 = c1 * c2;
      const float t13 = t12 * c3;
      const float t14 = t13 * c4;
      const float t15 = t14 * c5;                // c1*c2*c3*c4*c5
      lds[zr + 0] = 0.5f * (1.f + t15);          // wire0: bits 1..5
      lds[zr + 1] = 0.5f * (1.f + c0 * c1);      // wire1: bits 0,1
      lds[zr + 2] = 0.5f * (1.f + c0 * t12);     // wire2: bits 0..2
      lds[zr + 3] = 0.5f * (1.f + c0 * t13);     // wire3: bits 0..3
      lds[zr + 4] = 0.5f * (1.f + c0 * t14);     // wire4: bits 0..4
      lds[zr + 5] = 0.5f * (1.f + c0 * t15);     // wire5: bits 0..5
    }
    __syncthreads();

    // ============ GEMM2 (K=6, fused) + LSTM elementwise ============
    {
      float* orow = out + (size_t)t * (BATCH * HID);
#pragma unroll 1
      for (int bb = 0; bb < 16; ++bb) {
        const int b = bblk * 16 + bb;
        const float* gr = &lds[Z_OFF + b * Z_STRIDE];
        float G[24];
#pragma unroll
        for (int c4i = 0; c4i < 6; ++c4i) {
          v4f q = *(const v4f*)(gr + 4 * c4i);   // 16B-aligned (stride 96B)
          G[4 * c4i + 0] = q[0];
          G[4 * c4i + 1] = q[1];
          G[4 * c4i + 2] = q[2];
          G[4 * c4i + 3] = q[3];
        }
        float hv[4];
#pragma unroll
        for (int j = 0; j < 4; ++j) {
          float pf = bhv[j], pi_ = bhv[j], pg = bhv[j], po = bhv[j];
#pragma unroll
          for (int w = 0; w < 6; ++w) {
            const float whv = wh[j][w];
            pf += G[w] * whv;
            pi_ += G[6 + w] * whv;
            pg += G[12 + w] * whv;
            po += G[18 + w] * whv;
          }
          const float f = sigmoidf_(pf);
          const float ii = sigmoidf_(pi_);
          const float gg = tanhf(pg);
          const float oo = sigmoidf_(po);
          const size_t cidx = (size_t)b * HID + n0 + j;
          const float cprev = (t == 0) ? 0.f : cstate[cidx];
          const float cv = f * cprev + ii * gg;
          cstate[cidx] = cv;
          hv[j] = oo * tanhf(cv);
        }
        // h -> LDS (for next step's GEMM1) + global outputs[t]
        const int hb = H_OFF + b * H_STRIDE + n0;
        v2f h01; h01[0] = hv[0]; h01[1] = hv[1];
        v2f h23; h23[0] = hv[2]; h23[1] = hv[3];
        *(v2f*)&lds[hb] = h01;
        *(v2f*)&lds[hb + 2] = h23;
        v4f h4; h4[0] = hv[0]; h4[1] = hv[1]; h4[2] = hv[2]; h4[3] = hv[3];
        *(v4f*)&orow[(size_t)b * HID + n0] = h4;
      }
    }
    __syncthreads();
  }

  // ---- tail: hx = h(T-1) (from LDS), cx = c (from cstate) ----
  float* hx = out + (size_t)SEQ * BATCH * HID;
  float* cx = hx + BATCH * HID;
#pragma unroll 1
  for (int bb = 0; bb < 16; ++bb) {
    const int b = bblk * 16 + bb;
    const int hb = H_OFF + b * H_STRIDE + n0;
    v2f h01 = *(const v2f*)&lds[hb];
    v2f h23 = *(const v2f*)&lds[hb + 2];
    v4f h4; h4[0] = h01[0]; h4[1] = h01[1]; h4[2] = h23[0]; h4[3] = h23[1];
    *(v4f*)&hx[(size_t)b * HID + n0] = h4;
    const size_t cidx = (size_t)b * HID + n0;
    v4f c4;
#pragma unroll
    for (int j = 0; j < 4; ++j) c4[j] = cstate[cidx + j];
    *(v4f*)&cx[cidx] = c4;
  }
}

extern "C" void kernel_launch(void* const* d_in, const int* in_sizes, int n_in,
                              void* d_out, int out_size, void* d_ws,
                              size_t ws_size, hipStream_t stream) {
  (void)in_sizes; (void)n_in; (void)out_size; (void)ws_size;
  const float* inp = (const float*)d_in[0];
  const float* Wf  = (const float*)d_in[1];
  const float* bf  = (const float*)d_in[2];
  const float* Wi  = (const float*)d_in[3];
  const float* bi  = (const float*)d_in[4];
  const float* Wu  = (const float*)d_in[5];
  const float* bu  = (const float*)d_in[6];
  const float* Wo  = (const float*)d_in[7];
  const float* bo  = (const float*)d_in[8];
  const float* Wh  = (const float*)d_in[9];
  const float* bh  = (const float*)d_in[10];
  float* out = (float*)d_out;
  float* cstate = (float*)d_ws;  // BATCH*HID f32 cell state

  qlstm_persistent<<<dim3(1), dim3(1024), LDS_BYTES, stream>>>(
      inp, Wf, bf, Wi, bi, Wu, bu, Wo, bo, Wh, bh, out, cstate);
}